// PVSeg_22222160789827
// MI455X (gfx1250) — compile-verified
//
#include <hip/hip_runtime.h>
#include <hip/hip_bf16.h>

#define N_PTS   2000000
#define NUM_VOX 1200000
#define NCLS    20

typedef __attribute__((ext_vector_type(2))) float v2f;
typedef __attribute__((ext_vector_type(8))) float v8f;

__device__ __forceinline__ v8f wmma_k4(v2f a, v2f b, v8f c) {
    // D = A(16x4 f32) * B(4x16 f32) + C(16x16 f32)
    return __builtin_amdgcn_wmma_f32_16x16x4_f32(
        /*neg_a=*/false, a, /*neg_b=*/false, b,
        /*c_mod=*/(short)0, c, /*reuse_a=*/false, /*reuse_b=*/false);
}

__device__ __forceinline__ void lds_fence_wave() {
    __builtin_amdgcn_wave_barrier();
    asm volatile("s_wait_dscnt 0" ::: "memory");
    __builtin_amdgcn_wave_barrier();
}

// ---------------- kernel 1: zero voxel accumulators ----------------
__global__ void k_zero(float4* __restrict__ p, int n4) {
    int i = blockIdx.x * blockDim.x + threadIdx.x;
    if (i < n4) p[i] = make_float4(0.f, 0.f, 0.f, 0.f);
}

// ---------------- kernel 2: scatter-add per-point features ----------------
// seg[v*5 + {0,1,2}] = sum offset_xyz, +3 = sum intensity, +4 = count
__global__ void k_scatter(const float* __restrict__ lidar,
                          const float* __restrict__ vox,
                          const int*   __restrict__ inv,
                          float*       __restrict__ seg, int n) {
    int i = blockIdx.x * blockDim.x + threadIdx.x;
    if (i >= n) return;
    const float* r = lidar + (size_t)i * 5;
    float qx = r[1] / vox[0], qy = r[2] / vox[1], qz = r[3] / vox[2];
    float ox = floorf(qx) + 0.5f - qx;
    float oy = floorf(qy) + 0.5f - qy;
    float oz = floorf(qz) + 0.5f - qz;
    float* s = seg + (size_t)inv[i] * 5;
    atomicAdd(s + 0, ox);
    atomicAdd(s + 1, oy);
    atomicAdd(s + 2, oz);
    atomicAdd(s + 3, r[4]);
    atomicAdd(s + 4, 1.0f);
}

// ---------------- kernel 3: per-voxel average -> feat_sp (4 floats) ----------------
__global__ void k_voxel(const float* __restrict__ seg,
                        float*       __restrict__ fsb, int nv) {
    int v = blockIdx.x * blockDim.x + threadIdx.x;
    if (v >= nv) return;
    const float* s = seg + (size_t)v * 5;
    float rc = 1.0f / fmaxf(s[4], 1.0f);
    float4 o;
    o.x = s[0] * rc; o.y = s[1] * rc; o.z = s[2] * rc; o.w = s[3] * rc;
    ((float4*)fsb)[v] = o;
}

// ---------------- kernel 4: fused WMMA (vox MLP + point MLP + head) ----------------
// One wave per 16-point tile. All matmuls done transposed:
//   vox_featT = relu(W_vT @ feat_spT), p_featT = relu(W_pT @ xpT)  (16x16x4 WMMA)
//   outT      = W_oT(32x64, rows>=20 zero) @ featT(64x16)          (16 K-steps of 16x16x4)
__global__ __launch_bounds__(256) void k_wmma(const float* __restrict__ lidar,
                                              const float* __restrict__ vox,
                                              const float* __restrict__ Wv,
                                              const float* __restrict__ Wp,
                                              const float* __restrict__ Wo,
                                              const int*   __restrict__ inv,
                                              const float* __restrict__ fsb,
                                              float*       __restrict__ out,
                                              int n, int ntiles) {
    __shared__ float lds[8 * 1088];          // per-wave 64 channels x 17 stride
    const int lane = threadIdx.x & 31;
    const int wave = threadIdx.x >> 5;
    const int p    = lane & 15;              // point within tile / N index
    const int h    = lane >> 4;              // K-half selector
    const int wb   = wave * 1088;

    // ---- preload static A fragments (weights, transposed orientation) ----
    // A layout (16x4 f32): lane m = M row, v0 = K(2h), v1 = K(2h+1)
    v2f awv[2], awp[2];
#pragma unroll
    for (int t = 0; t < 2; ++t) {
        int ch = t * 16 + p;                 // output channel = M row
        awv[t].x = Wv[(2 * h + 0) * 32 + ch];
        awv[t].y = Wv[(2 * h + 1) * 32 + ch];
        awp[t].x = Wp[(2 * h + 0) * 32 + ch];
        awp[t].y = Wp[(2 * h + 1) * 32 + ch];
    }
    v2f awo[2][16];
#pragma unroll
    for (int m = 0; m < 2; ++m) {
        int cls = m * 16 + p;
        bool ok = (cls < NCLS);
#pragma unroll
        for (int k = 0; k < 16; ++k) {
            int K = 4 * k + 2 * h;
            awo[m][k].x = ok ? Wo[(size_t)K * NCLS + cls] : 0.0f;
            awo[m][k].y = ok ? Wo[(size_t)(K + 1) * NCLS + cls] : 0.0f;
        }
    }
    const float vxa = vox[2 * h];            // divisor for feature k = 2h   (x or z)
    const float vxb = vox[1];                // divisor for feature k = 1    (y, h==0 only)

    const int gw = blockIdx.x * 8 + wave;
    const int nw = gridDim.x * 8;

    for (int t = gw; t < ntiles; t += nw) {
        int pi = t * 16 + p;
        if (pi >= n) pi = n - 1;             // clamp: keeps EXEC all-1s for WMMA

        // ---- build B fragments (B layout: lane n = point col, v0=K(2h), v1=K(2h+1)) ----
        int vix = inv[pi];
        float2 f2 = *(const float2*)(fsb + (size_t)vix * 4 + 2 * h);
        v2f bfs; bfs.x = f2.x; bfs.y = f2.y;

        float l1 = lidar[(size_t)pi * 5 + 1 + 2 * h];
        float l2 = lidar[(size_t)pi * 5 + 2 + 2 * h];
        v2f bxp;
        { float q = l1 / vxa; bxp.x = floorf(q) + 0.5f - q; }
        if (h) { bxp.y = l2; }               // k=3: raw intensity
        else   { float q = l2 / vxb; bxp.y = floorf(q) + 0.5f - q; }

        // ---- layer 1: two 16x32 transposed MLPs, relu ----
        v8f z = {};
        v8f dv0 = wmma_k4(awv[0], bfs, z);
        v8f dv1 = wmma_k4(awv[1], bfs, z);
        v8f dp0 = wmma_k4(awp[0], bxp, z);
        v8f dp1 = wmma_k4(awp[1], bxp, z);
#pragma unroll
        for (int r = 0; r < 8; ++r) {
            dv0[r] = fmaxf(dv0[r], 0.0f);
            dv1[r] = fmaxf(dv1[r], 0.0f);
            dp0[r] = fmaxf(dp0[r], 0.0f);
            dp1[r] = fmaxf(dp1[r], 0.0f);
        }

        // ---- stage featT (64 ch x 16 pts) through LDS: lds[c*17 + p] ----
#pragma unroll
        for (int r = 0; r < 8; ++r) {
            int c = r + 8 * h;
            lds[wb + (c +  0) * 17 + p] = dv0[r];
            lds[wb + (c + 16) * 17 + p] = dv1[r];
            lds[wb + (c + 32) * 17 + p] = dp0[r];
            lds[wb + (c + 48) * 17 + p] = dp1[r];
        }
        lds_fence_wave();

        // ---- head: outT = W_oT @ featT, 16 K-steps of 16x16x4 ----
        v8f acc0 = {}, acc1 = {};
#pragma unroll
        for (int k = 0; k < 16; ++k) {
            int c0 = 4 * k + 2 * h;
            v2f b;
            b.x = lds[wb + c0 * 17 + p];
            b.y = lds[wb + (c0 + 1) * 17 + p];
            acc0 = wmma_k4(awo[0][k], b, acc0);
            acc1 = wmma_k4(awo[1][k], b, acc1);
        }

        // ---- store: D lane = point, VGPR r = class r + 8h; classes contiguous ----
        float* orow = out + (size_t)pi * NCLS;
        union { v8f v; float4 q[2]; } u0, u1;
        u0.v = acc0; u1.v = acc1;
        if (h == 0) {
            *(float4*)(orow + 0)  = u0.q[0];   // classes 0-3
            *(float4*)(orow + 4)  = u0.q[1];   // classes 4-7
            *(float4*)(orow + 16) = u1.q[0];   // classes 16-19
        } else {
            *(float4*)(orow + 8)  = u0.q[0];   // classes 8-11
            *(float4*)(orow + 12) = u0.q[1];   // classes 12-15
        }
    }
}

extern "C" void kernel_launch(void* const* d_in, const int* in_sizes, int n_in,
                              void* d_out, int out_size, void* d_ws, size_t ws_size,
                              hipStream_t stream) {
    const float* lidar = (const float*)d_in[0];   // (N,5)
    const float* vox   = (const float*)d_in[1];   // (3,)
    const float* Wv    = (const float*)d_in[2];   // (4,32)
    const float* Wp    = (const float*)d_in[3];   // (4,32)
    const float* Wo    = (const float*)d_in[4];   // (64,20)
    const int*   inv   = (const int*)d_in[5];     // (N,)
    float* out = (float*)d_out;

    const int n = in_sizes[0] / 5;
    float* seg = (float*)d_ws;                       // NUM_VOX * 5 accumulators
    float* fsb = seg + (size_t)NUM_VOX * 5;          // NUM_VOX * 4 feat_sp

    const int n4 = (NUM_VOX * 5) / 4;                // 6,000,000 floats -> float4
    k_zero<<<(n4 + 255) / 256, 256, 0, stream>>>((float4*)seg, n4);
    k_scatter<<<(n + 255) / 256, 256, 0, stream>>>(lidar, vox, inv, seg, n);
    k_voxel<<<(NUM_VOX + 255) / 256, 256, 0, stream>>>(seg, fsb, NUM_VOX);

    const int ntiles = (n + 15) / 16;
    k_wmma<<<1024, 256, 0, stream>>>(lidar, vox, Wv, Wp, Wo, inv, fsb, out, n, ntiles);
}